// PixelDSNTDoubleEval_55035710931298
// MI455X (gfx1250) — compile-verified
//
#include <hip/hip_runtime.h>
#include <math.h>

#ifndef __has_builtin
#define __has_builtin(x) 0
#endif

// Problem geometry (fixed by reference setup_inputs)
#define B_SZ 32
#define C_SZ 2
#define H_SZ 512
#define W_SZ 512
#define PLANE (H_SZ * W_SZ)                 // 262144 elements per (b,c) plane
#define NPLANES (B_SZ * C_SZ)               // 64
#define CHUNKS_PER_PLANE 16
#define CHUNK (PLANE / CHUNKS_PER_PLANE)    // 16384 floats per block
#define NTHREADS 256
#define TILE (NTHREADS * 4)                 // 1024 floats per tile (float4/lane)
#define TILES (CHUNK / TILE)                // 16 tiles per block
#define DEPTH 4                             // async pipeline depth (buffers)
#define NBLOCKS (NPLANES * CHUNKS_PER_PLANE) // 1024

struct Partial {
  float m, s, sx, sy, tv;
  int   ti;
  int   pad0, pad1;                          // 32-byte stride
};

#if __has_builtin(__builtin_amdgcn_global_load_async_to_lds_b128)
#define HAVE_ASYNC_LDS 1
#else
#define HAVE_ASYNC_LDS 0
#endif

#if __has_builtin(__builtin_amdgcn_s_wait_asynccnt)
#define ASYNC_WAIT(N) __builtin_amdgcn_s_wait_asynccnt(N)
#else
#define ASYNC_WAIT(N) asm volatile("s_wait_asynccnt " #N ::: "memory")
#endif

__device__ __forceinline__ void wait_async_le(int n) {
#if HAVE_ASYNC_LDS
  if (n <= 0)      { ASYNC_WAIT(0); }
  else if (n <= 2) { ASYNC_WAIT(2); }
  else if (n <= 4) { ASYNC_WAIT(4); }
  else             { ASYNC_WAIT(6); }
#else
  (void)n;
#endif
}

#if HAVE_ASYNC_LDS
// The builtin takes pointers to a 4 x i32 vector in AS1 (global) / AS3 (LDS).
typedef __attribute__((__vector_size__(4 * sizeof(int)))) int v4i;
typedef __attribute__((address_space(1))) v4i* gptr_v4i;
typedef __attribute__((address_space(3))) v4i* lptr_v4i;

__device__ __forceinline__ void async_copy16(const float* gsrc, float* ldst) {
  __builtin_amdgcn_global_load_async_to_lds_b128(
      (gptr_v4i)gsrc, (lptr_v4i)ldst, 0, 0);
}
#endif

// Online-softmax accumulate of 4 consecutive elements (same row => same wy).
__device__ __forceinline__ void accum4(float& m, float& s, float& sx, float& sy,
                                       float& tv, int& ti,
                                       float4 vi, float4 vt, int e0) {
  const int   hrow = e0 >> 9;          // H index within plane (W=512)
  const int   wcol = e0 & (W_SZ - 1);  // W index of first element
  const float wx0  = (float)(wcol + 1);
  const float wy   = (float)(hrow + 1);

  float lm = fmaxf(fmaxf(vi.x, vi.y), fmaxf(vi.z, vi.w));
  float mn = fmaxf(m, lm);
  float r  = __expf(m - mn);           // rescale old accumulators
  float e_0 = __expf(vi.x - mn);
  float e_1 = __expf(vi.y - mn);
  float e_2 = __expf(vi.z - mn);
  float e_3 = __expf(vi.w - mn);
  float es = e_0 + e_1 + e_2 + e_3;
  float ew = e_1 + 2.0f * e_2 + 3.0f * e_3;  // Σ e_j * j
  s  = fmaf(s,  r, es);
  sx = fmaf(sx, r, fmaf(es, wx0, ew));       // Σ e_j*(wx0+j)
  sy = fmaf(sy, r, es * wy);
  m  = mn;

  // first-occurrence argmax (ascending index, strict > keeps first)
  if (vt.x > tv) { tv = vt.x; ti = e0;     }
  if (vt.y > tv) { tv = vt.y; ti = e0 + 1; }
  if (vt.z > tv) { tv = vt.z; ti = e0 + 2; }
  if (vt.w > tv) { tv = vt.w; ti = e0 + 3; }
}

__global__ __launch_bounds__(NTHREADS)
void dsnt_partial_kernel(const float* __restrict__ in,
                         const float* __restrict__ tg,
                         Partial* __restrict__ ws) {
  __shared__ float red_m[NTHREADS];
  __shared__ float red_s[NTHREADS];
  __shared__ float red_x[NTHREADS];
  __shared__ float red_y[NTHREADS];
  __shared__ float red_tv[NTHREADS];
  __shared__ int   red_ti[NTHREADS];

  const int tid   = threadIdx.x;
  const int blk   = blockIdx.x;
  const int plane = blk / CHUNKS_PER_PLANE;
  const int chunk = blk - plane * CHUNKS_PER_PLANE;
  const int chunkBase = chunk * CHUNK;                 // element offset in plane
  const size_t gbase  = (size_t)plane * PLANE + (size_t)chunkBase;

  const float* gin = in + gbase;
  const float* gtg = tg + gbase;

  float m = -__builtin_inff(), s = 0.0f, sx = 0.0f, sy = 0.0f;
  float tv = -__builtin_inff();
  int   ti = 0x7fffffff;

#if HAVE_ASYNC_LDS
  __shared__ __align__(16) float stg_in[DEPTH][TILE];
  __shared__ __align__(16) float stg_tg[DEPTH][TILE];

  int issued = 0;
  #pragma unroll
  for (int p = 0; p < DEPTH; ++p) {
    const int b = p & (DEPTH - 1);
    async_copy16(gin + p * TILE + tid * 4, &stg_in[b][tid * 4]);
    async_copy16(gtg + p * TILE + tid * 4, &stg_tg[b][tid * 4]);
    ++issued;
  }

  #pragma unroll
  for (int t = 0; t < TILES; ++t) {
    const int rem = TILES - 1 - t;
    wait_async_le(2 * (rem < (DEPTH - 1) ? rem : (DEPTH - 1)));  // oldest tile landed

    const int b = t & (DEPTH - 1);
    float4 vi = *(const float4*)(&stg_in[b][tid * 4]);
    float4 vt = *(const float4*)(&stg_tg[b][tid * 4]);
    // Ensure the ds_loads above have returned before the async engine may
    // overwrite this buffer with tile t+DEPTH (issued just below).
    asm volatile("s_wait_dscnt 0" ::: "memory");

    if (issued < TILES) {
      const int nb = issued & (DEPTH - 1);
      async_copy16(gin + issued * TILE + tid * 4, &stg_in[nb][tid * 4]);
      async_copy16(gtg + issued * TILE + tid * 4, &stg_tg[nb][tid * 4]);
      ++issued;
    }

    accum4(m, s, sx, sy, tv, ti, vi, vt, chunkBase + t * TILE + tid * 4);
  }
#else
  #pragma unroll 4
  for (int t = 0; t < TILES; ++t) {
    float4 vi = *(const float4*)(gin + t * TILE + tid * 4);
    float4 vt = *(const float4*)(gtg + t * TILE + tid * 4);
    accum4(m, s, sx, sy, tv, ti, vi, vt, chunkBase + t * TILE + tid * 4);
  }
#endif

  // Block reduction (online-softmax merge + argmax merge).
  red_m[tid] = m;  red_s[tid] = s;  red_x[tid] = sx;  red_y[tid] = sy;
  red_tv[tid] = tv; red_ti[tid] = ti;
  __syncthreads();
  for (int off = NTHREADS / 2; off > 0; off >>= 1) {
    if (tid < off) {
      float m1 = red_m[tid],        s1 = red_s[tid];
      float x1 = red_x[tid],        y1 = red_y[tid];
      float m2 = red_m[tid + off],  s2 = red_s[tid + off];
      float x2 = red_x[tid + off],  y2 = red_y[tid + off];
      float mn = fmaxf(m1, m2);
      float r1 = __expf(m1 - mn);
      float r2 = __expf(m2 - mn);
      red_m[tid] = mn;
      red_s[tid] = fmaf(s1, r1, s2 * r2);
      red_x[tid] = fmaf(x1, r1, x2 * r2);
      red_y[tid] = fmaf(y1, r1, y2 * r2);
      float tv2 = red_tv[tid + off]; int ti2 = red_ti[tid + off];
      if (tv2 > red_tv[tid] || (tv2 == red_tv[tid] && ti2 < red_ti[tid])) {
        red_tv[tid] = tv2; red_ti[tid] = ti2;
      }
    }
    __syncthreads();
  }

  if (tid == 0) {
    Partial p = { red_m[0], red_s[0], red_x[0], red_y[0],
                  red_tv[0], red_ti[0], 0, 0 };
    ws[blk] = p;
  }
}

__global__ __launch_bounds__(64)
void dsnt_final_kernel(const Partial* __restrict__ ws, float* __restrict__ out) {
  __shared__ float ed_s[NPLANES];
  const int p = threadIdx.x;  // one plane per thread (64 planes)

  if (p < NPLANES) {
    float m = -__builtin_inff(), s = 0.0f, sx = 0.0f, sy = 0.0f;
    float tv = -__builtin_inff();
    int   ti = 0x7fffffff;
    for (int c = 0; c < CHUNKS_PER_PLANE; ++c) {
      Partial q = ws[p * CHUNKS_PER_PLANE + c];
      float mn = fmaxf(m, q.m);
      float r1 = __expf(m - mn);
      float r2 = __expf(q.m - mn);
      s  = fmaf(s,  r1, q.s  * r2);
      sx = fmaf(sx, r1, q.sx * r2);
      sy = fmaf(sy, r1, q.sy * r2);
      m  = mn;
      if (q.tv > tv || (q.tv == tv && q.ti < ti)) { tv = q.tv; ti = q.ti; }
    }
    float px = sx / s;                       // soft-argmax x in pixels
    float py = sy / s;                       // soft-argmax y in pixels
    float tx = (float)((ti & (W_SZ - 1)) + 1);
    float ty = (float)((ti >> 9) + 1);
    float dx = tx - px, dy = ty - py;
    ed_s[p] = sqrtf(dx * dx + dy * dy);
  }
  __syncthreads();

  if (p == 0) {
    float si = 0.0f, ss = 0.0f;
    for (int q = 0; q < NPLANES; ++q) {
      if ((q % C_SZ) == 0) si += ed_s[q];    // channel 0 -> inferior
      else                 ss += ed_s[q];    // channels 1.. -> superior
    }
    const float denom = (float)B_SZ;
    out[0] = si / denom;
    out[1] = ss / denom;
    out[2] = (si + ss) / denom;
  }
}

extern "C" void kernel_launch(void* const* d_in, const int* in_sizes, int n_in,
                              void* d_out, int out_size, void* d_ws, size_t ws_size,
                              hipStream_t stream) {
  (void)in_sizes; (void)n_in; (void)out_size; (void)ws_size;
  const float* in = (const float*)d_in[0];
  const float* tg = (const float*)d_in[1];
  float*  out = (float*)d_out;
  Partial* ws = (Partial*)d_ws;   // 1024 * 32 B = 32 KB of scratch

  dsnt_partial_kernel<<<NBLOCKS, NTHREADS, 0, stream>>>(in, tg, ws);
  dsnt_final_kernel<<<1, 64, 0, stream>>>(ws, out);
}